// MSRRBlock_13383118094546
// MI455X (gfx1250) — compile-verified
//
#include <hip/hip_runtime.h>
#include <math.h>

typedef float v2f __attribute__((ext_vector_type(2)));
typedef float v4f __attribute__((ext_vector_type(4)));
typedef float v8f __attribute__((ext_vector_type(8)));
typedef unsigned int v4u __attribute__((ext_vector_type(4)));
typedef int v4i __attribute__((ext_vector_type(4)));
typedef int v8i __attribute__((ext_vector_type(8)));

#define D_DIM 768
#define R_DIM 3
#define NE 4
#define B_DIM 8
#define N_DIM 16384
#define HW 128

// Use the Tensor Data Mover for the proj_down x-tile staging when available.
#ifndef USE_TDM
#if defined(__has_builtin)
#if __has_builtin(__builtin_amdgcn_tensor_load_to_lds)
#define USE_TDM 1
#else
#define USE_TDM 0
#endif
#else
#define USE_TDM 0
#endif
#endif

// ---------------- workspace layout (floats) ----------------
#define WS_XR    0          // B*N*R = 393216
#define WS_MIXED 393216     // B*N*R = 393216
#define WS_XA    786432     // 24 (pad 32)
#define WS_G     786464     // 32

// ---------------- kernel 0: zero the mean accumulator ----------------
__global__ void init_kernel(float* xa_sum) {
    if (threadIdx.x < B_DIM * R_DIM) xa_sum[threadIdx.x] = 0.0f;
}

#if USE_TDM
// Issue one TDM 2D tile load: 128 rows x 64 floats from row-major x (row stride
// 768 floats) into LDS at ldsOff, padding +4 DWORDs after every 64 DWORDs so the
// LDS tile lands with row stride 68 (bank-conflict-free WMMA A-fragment reads).
__device__ inline void tdm_load_tile(unsigned ldsOff, const float* gsrc) {
    unsigned long long ga = (unsigned long long)(size_t)gsrc;
    v4u g0;
    g0[0] = 1u;                                   // count=1, is_restore=0, gather=0
    g0[1] = ldsOff;                               // lds_addr (byte offset)
    g0[2] = (unsigned)(ga & 0xffffffffull);       // global_addr[31:0]
    g0[3] = (unsigned)((ga >> 32) & 0x01ffffffull) | 0x80000000u; // addr[56:32] | type=2
    v8i g1;
    g1[0] = (int)((2u << 16)      // data_size = 4 bytes
                | (1u << 20)      // pad_enable
                | (5u << 22)      // pad_interval: 64 DWORDs
                | (3u << 25));    // pad_amount : 4 DWORDs
    g1[1] = (int)(768u << 16);    // tensor_dim0 = 768 (low 16)
    g1[2] = 0;                    // tensor_dim0 hi / tensor_dim1 lo
    g1[3] = (int)(2u | (64u << 16));   // tensor_dim1 = 131072 (hi16=2); tile_dim0 = 64
    g1[4] = (int)128u;            // tile_dim1 = 128; tile_dim2 = 0
    g1[5] = (int)768u;            // tensor_dim0_stride = 768
    g1[6] = 0;
    g1[7] = 0;
    v4i z4 = {0, 0, 0, 0};
#if __clang_major__ >= 23
    v8i z8 = {0, 0, 0, 0, 0, 0, 0, 0};
    __builtin_amdgcn_tensor_load_to_lds(g0, g1, z4, z4, z8, 0);
#else
    __builtin_amdgcn_tensor_load_to_lds(g0, g1, z4, z4, 0);
#endif
}
#endif

// ---------------- kernel 1: xr = x @ Wd + bd,  xa += sums ----------------
// 1024 blocks * 256 threads. Block: 128 tokens, wave: 16 tokens.
// K-chunked (64) staging of x into LDS (TDM or b128 cooperative loads),
// then V_WMMA_F32_16X16X4_F32 accumulation (192 WMMA per wave).
__global__ void __launch_bounds__(256) proj_down_kernel(
    const float* __restrict__ x, const float* __restrict__ Wd,
    const float* __restrict__ bd, float* __restrict__ xr,
    float* __restrict__ xa_sum)
{
    __shared__ float sA[128 * 68];        // 128 tokens x 64 K, stride 68
    __shared__ float sW[D_DIM * R_DIM];   // whole Wd (768x3)
    __shared__ float sPart[16 * R_DIM];

    const int tid  = threadIdx.x;
    const int wave = tid >> 5;
    const int lane = tid & 31;
    const int L    = lane & 15;
    const int half = lane >> 4;
    const size_t tokBlk = (size_t)blockIdx.x * 128;

    for (int i = tid; i < D_DIM * R_DIM; i += 256) sW[i] = Wd[i];

    v8f c = {};
    const int lt = wave * 16 + L;

    for (int kc = 0; kc < D_DIM; kc += 64) {
        __syncthreads();   // previous tile fully consumed
#if USE_TDM
        if (wave == 0) {
            tdm_load_tile((unsigned)(size_t)&sA[0], x + tokBlk * D_DIM + kc);
            __builtin_amdgcn_s_wait_tensorcnt(0);
        }
#else
        // coalesced b128 staging: consecutive tids -> consecutive 16B within a row
        for (int i = tid; i < 128 * 16; i += 256) {
            int row = i >> 4, c4 = (i & 15) * 4;
            *(v4f*)&sA[row * 68 + c4] =
                *(const v4f*)&x[(tokBlk + row) * D_DIM + kc + c4];
        }
#endif
        __syncthreads();   // tile visible to all waves

        #pragma unroll
        for (int kk = 0; kk < 64; kk += 4) {
            // A fragment: lanes 0-15 hold (K0,K1), lanes 16-31 hold (K2,K3)
            v2f a = *(const v2f*)&sA[lt * 68 + kk + half * 2];
            // B fragment: column n = L; only n<3 are real columns of Wd
            v2f bfrag; bfrag.x = 0.0f; bfrag.y = 0.0f;
            if (L < R_DIM) {
                int k0 = kc + kk + half * 2;
                bfrag.x = sW[k0 * R_DIM + L];
                bfrag.y = sW[(k0 + 1) * R_DIM + L];
            }
            c = __builtin_amdgcn_wmma_f32_16x16x4_f32(
                    false, a, false, bfrag, (short)0, c, false, false);
        }
    }

    // C layout: vgpr j -> row (j + half*8), column = L
    if (L < R_DIM) {
        float bdr = bd[L];
        size_t tok0 = tokBlk + (size_t)wave * 16 + half * 8;
        float sum8 = 0.0f;
        #pragma unroll
        for (int j = 0; j < 8; ++j) {
            float v = c[j] + bdr;
            xr[(tok0 + j) * R_DIM + L] = v;
            sum8 += v;
        }
        sPart[(wave * 2 + half) * R_DIM + L] = sum8;
    }
    __syncthreads();
    if (tid < R_DIM) {
        float t = 0.0f;
        #pragma unroll
        for (int i = 0; i < 16; ++i) t += sPart[i * R_DIM + tid];
        int b = blockIdx.x >> 7;  // 128 blocks per batch
        atomicAdd(&xa_sum[b * R_DIM + tid], t);
    }
}

// ---------------- kernel 2: gating -> G (B x NE) ----------------
__global__ void gating_kernel(const float* __restrict__ xa_sum,
                              const float* __restrict__ noise,
                              const float* __restrict__ Wg,
                              const float* __restrict__ Wn,
                              float* __restrict__ G)
{
    int b = threadIdx.x;
    if (b >= B_DIM) return;
    const float inv_n = 1.0f / (float)N_DIM;
    float xa0 = xa_sum[b * R_DIM + 0] * inv_n;
    float xa1 = xa_sum[b * R_DIM + 1] * inv_n;
    float xa2 = xa_sum[b * R_DIM + 2] * inv_n;

    float hlog[NE];
    #pragma unroll
    for (int e = 0; e < NE; ++e) {
        float hg = xa0 * Wg[e] + xa1 * Wg[NE + e] + xa2 * Wg[2 * NE + e];
        float hn = xa0 * Wn[e] + xa1 * Wn[NE + e] + xa2 * Wn[2 * NE + e];
        float sp = (hn > 20.0f) ? hn : log1pf(expf(hn));   // softplus
        hlog[e] = hg + noise[b * NE + e] * sp;
    }
    int i1 = 0;
    #pragma unroll
    for (int e = 1; e < NE; ++e) if (hlog[e] > hlog[i1]) i1 = e;
    int i2 = -1;
    #pragma unroll
    for (int e = 0; e < NE; ++e)
        if (e != i1 && (i2 < 0 || hlog[e] > hlog[i2])) i2 = e;

    float m  = hlog[i1];
    float e1 = expf(hlog[i1] - m);
    float e2 = expf(hlog[i2] - m);
    float inv = 1.0f / (e1 + e2);
    #pragma unroll
    for (int e = 0; e < NE; ++e)
        G[b * NE + e] = (e == i1) ? e1 * inv : ((e == i2) ? e2 * inv : 0.0f);
}

// ---------------- kernel 3: experts (resize -> dwconv -> resize) ----------------
__device__ inline void lin_coef(float s, int n, int& i0, int& i1, float& w1) {
    float f = floorf(s);
    w1 = s - f;
    int i = (int)f;
    i0 = min(max(i, 0), n - 1);
    i1 = min(max(i + 1, 0), n - 1);
}

__global__ void __launch_bounds__(256) experts_kernel(
    const float* __restrict__ xr, const float* __restrict__ G,
    const float* __restrict__ dwk, const float* __restrict__ dwb,
    float* __restrict__ mixed)
{
    __shared__ float s_down[64 * 64];
    __shared__ float s_conv[64 * 64];

    const int b = blockIdx.x / R_DIM;
    const int ch = blockIdx.x % R_DIM;
    const int tid = threadIdx.x;

    float kw[9];
    #pragma unroll
    for (int i = 0; i < 9; ++i) kw[i] = dwk[ch * 9 + i];
    const float kb = dwb[ch];
    float g[NE];
    #pragma unroll
    for (int j = 0; j < NE; ++j) g[j] = G[b * NE + j];

    const size_t featBase = (size_t)b * N_DIM;
    #define FEAT(i, k) xr[(featBase + (size_t)(i) * HW + (k)) * R_DIM + ch]

    // ---- scale 1.0: conv on full-res plane, initialize mixed ----
    for (int p = tid; p < HW * HW; p += 256) {
        int i = p >> 7, k = p & 127;
        float acc = kb;
        #pragma unroll
        for (int di = -1; di <= 1; ++di) {
            #pragma unroll
            for (int dk = -1; dk <= 1; ++dk) {
                int ii = i + di, kk = k + dk;
                float v = (ii >= 0 && ii < HW && kk >= 0 && kk < HW) ? FEAT(ii, kk) : 0.0f;
                acc += kw[(di + 1) * 3 + (dk + 1)] * v;
            }
        }
        mixed[(featBase + p) * R_DIM + ch] = g[0] * acc;
    }

    // ---- scales 0.5, 0.25, 0.125 ----
    for (int j = 1; j < NE; ++j) {
        const int hs = HW >> j;
        __syncthreads();
        {   // downsample feat -> s_down (half-pixel bilinear, clamped)
            float sf = (float)HW / (float)hs;
            for (int p = tid; p < hs * hs; p += 256) {
                int oi = p / hs, ok = p % hs;
                int i0, i1, k0, k1; float wi, wk;
                lin_coef((oi + 0.5f) * sf - 0.5f, HW, i0, i1, wi);
                lin_coef((ok + 0.5f) * sf - 0.5f, HW, k0, k1, wk);
                float v00 = FEAT(i0, k0), v01 = FEAT(i0, k1);
                float v10 = FEAT(i1, k0), v11 = FEAT(i1, k1);
                s_down[p] = (1.0f - wi) * ((1.0f - wk) * v00 + wk * v01)
                          +          wi * ((1.0f - wk) * v10 + wk * v11);
            }
        }
        __syncthreads();
        // depthwise 3x3, zero pad
        for (int p = tid; p < hs * hs; p += 256) {
            int i = p / hs, k = p % hs;
            float acc = kb;
            #pragma unroll
            for (int di = -1; di <= 1; ++di) {
                #pragma unroll
                for (int dk = -1; dk <= 1; ++dk) {
                    int ii = i + di, kk = k + dk;
                    float v = (ii >= 0 && ii < hs && kk >= 0 && kk < hs)
                                  ? s_down[ii * hs + kk] : 0.0f;
                    acc += kw[(di + 1) * 3 + (dk + 1)] * v;
                }
            }
            s_conv[p] = acc;
        }
        __syncthreads();
        {   // upsample to 128x128, accumulate (same thread owns same pixels)
            float sf = (float)hs / (float)HW;
            for (int p = tid; p < HW * HW; p += 256) {
                int i = p >> 7, k = p & 127;
                int i0, i1, k0, k1; float wi, wk;
                lin_coef((i + 0.5f) * sf - 0.5f, hs, i0, i1, wi);
                lin_coef((k + 0.5f) * sf - 0.5f, hs, k0, k1, wk);
                float v00 = s_conv[i0 * hs + k0], v01 = s_conv[i0 * hs + k1];
                float v10 = s_conv[i1 * hs + k0], v11 = s_conv[i1 * hs + k1];
                float val = (1.0f - wi) * ((1.0f - wk) * v00 + wk * v01)
                          +          wi * ((1.0f - wk) * v10 + wk * v11);
                mixed[(featBase + p) * R_DIM + ch] += g[j] * val;
            }
        }
    }
    #undef FEAT
}

// ---------------- kernel 4: out = x + mixed @ Wu + bu ----------------
// 1024 blocks * 256 threads; each wave = one 16-token tile. Per 64-column
// group: 4 WMMA -> scatter C to per-wave LDS tile -> read back row-major
// -> fully-coalesced b128 streaming of x/out (2x 256B segments per instr).
__global__ void __launch_bounds__(256) proj_up_kernel(
    const float* __restrict__ x, const float* __restrict__ mixed,
    const float* __restrict__ Wu, const float* __restrict__ bu,
    float* __restrict__ out)
{
    __shared__ float sT[8 * 16 * 68];   // per-wave 16 rows x 68 (16B-aligned pad)

    const int tid  = threadIdx.x;
    const int wave = tid >> 5;
    const int lane = tid & 31;
    const int L    = lane & 15;
    const int half = lane >> 4;
    const size_t tok0 = ((size_t)blockIdx.x * 8 + wave) * 16;
    float* tw = &sT[wave * 16 * 68];

    // A fragment: K = {r0, r1} for lanes 0-15, {r2, 0} for lanes 16-31
    v2f a;
    {
        const float* mr = mixed + (tok0 + L) * R_DIM;
        if (half == 0) { a.x = mr[0]; a.y = mr[1]; }
        else           { a.x = mr[2]; a.y = 0.0f; }
    }

    const int colL = 4 * L;   // readback column within the 64-wide group

    for (int grp = 0; grp < D_DIM / 64; ++grp) {
        const int d0 = grp * 64;
        if (grp + 1 < D_DIM / 64)   // hint next x group into cache
            __builtin_prefetch(&x[(tok0 + half) * D_DIM + d0 + 64 + colL], 0, 0);

        #pragma unroll
        for (int g = 0; g < 4; ++g) {
            const int dg = d0 + g * 16;
            v2f bfrag;
            if (half == 0) { bfrag.x = Wu[dg + L];             bfrag.y = Wu[D_DIM + dg + L]; }
            else           { bfrag.x = Wu[2 * D_DIM + dg + L]; bfrag.y = 0.0f; }

            v8f c = {};
            c = __builtin_amdgcn_wmma_f32_16x16x4_f32(
                    false, a, false, bfrag, (short)0, c, false, false);

            // scatter C: row (j+half*8), local col g*16+L  (conflict-free banks)
            #pragma unroll
            for (int j = 0; j < 8; ++j)
                tw[(j + half * 8) * 68 + g * 16 + L] = c[j];
        }

        // read back row-major and stream: lanes 0-15 -> row 2q (256B contiguous),
        // lanes 16-31 -> row 2q+1 (256B contiguous)
        v4f bu4 = *(const v4f*)&bu[d0 + colL];
        #pragma unroll
        for (int q = 0; q < 8; ++q) {
            const int row = q * 2 + half;
            const size_t idx = (tok0 + row) * D_DIM + d0 + colL;
            v4f xv = *(const v4f*)&x[idx];
            v4f yv = *(const v4f*)&tw[row * 68 + colL];
            v4f ov = xv + yv + bu4;
            *(v4f*)&out[idx] = ov;
        }
    }
}

// ---------------- launcher ----------------
extern "C" void kernel_launch(void* const* d_in, const int* in_sizes, int n_in,
                              void* d_out, int out_size, void* d_ws, size_t ws_size,
                              hipStream_t stream)
{
    const float* x     = (const float*)d_in[0];
    const float* noise = (const float*)d_in[1];
    const float* Wd    = (const float*)d_in[2];
    const float* bd    = (const float*)d_in[3];
    const float* Wu    = (const float*)d_in[4];
    const float* bu    = (const float*)d_in[5];
    const float* Wg    = (const float*)d_in[6];
    const float* Wn    = (const float*)d_in[7];
    const float* dwk   = (const float*)d_in[8];
    const float* dwb   = (const float*)d_in[9];
    float* out = (float*)d_out;

    float* ws       = (float*)d_ws;
    float* ws_xr    = ws + WS_XR;
    float* ws_mixed = ws + WS_MIXED;
    float* ws_xa    = ws + WS_XA;
    float* ws_G     = ws + WS_G;

    init_kernel<<<1, 32, 0, stream>>>(ws_xa);
    proj_down_kernel<<<1024, 256, 0, stream>>>(x, Wd, bd, ws_xr, ws_xa);
    gating_kernel<<<1, 32, 0, stream>>>(ws_xa, noise, Wg, Wn, ws_G);
    experts_kernel<<<B_DIM * R_DIM, 256, 0, stream>>>(ws_xr, ws_G, dwk, dwb, ws_mixed);
    proj_up_kernel<<<1024, 256, 0, stream>>>(x, ws_mixed, Wu, bu, out);
}